// P_Selective_Shuffle_62113817035264
// MI455X (gfx1250) — compile-verified
//
#include <hip/hip_runtime.h>
#include <hip/hip_bf16.h>

// Selective patch shuffle: out patch i = in patch perm[b, i], 16x16 patches,
// B=64, C=3, H=W=512, fp32.  Pure data movement: ~403 MB total traffic ->
// ~17 us roofline at 23.3 TB/s.  Zero FLOPs, so no WMMA; optimal path is
// b128 vector loads/stores with all three channel loads in flight before the
// stores (split s_wait_loadcnt on gfx1250).

namespace {
constexpr unsigned kB = 64;
constexpr unsigned kC = 3;
constexpr unsigned kH = 512;
constexpr unsigned kW = 512;
constexpr unsigned kPatch = 16;
constexpr unsigned kNH = kW / kPatch;          // 32 patches across
constexpr unsigned kNPatch = (kH / kPatch) * kNH;  // 1024 patches / image
constexpr unsigned kPlane = kH * kW;           // elements per channel plane
constexpr unsigned kVecPerRow = kW / 4;        // 128 float4 per row
constexpr unsigned kThreadsPerImage = kH * kVecPerRow; // 65536
}

__global__ __launch_bounds__(256) void patch_shuffle_kernel(
    const float* __restrict__ img,   // [B, C, H, W]
    const int*   __restrict__ perm,  // [B, 1024]
    float*       __restrict__ out)   // [B, C, H, W]
{
    const unsigned tid = blockIdx.x * 256u + threadIdx.x;

    // Decompose: tid = ((b * H) + y) * (W/4) + x4   (all power-of-two strides)
    const unsigned x  = (tid & (kVecPerRow - 1u)) << 2;  // 0..508 step 4
    const unsigned y  = (tid >> 7) & (kH - 1u);          // 0..511
    const unsigned b  = tid >> 16;                       // 0..63

    // Output patch index (row-major over 32x32 patch grid)
    const unsigned pi = ((y >> 4) << 5) | (x >> 4);
    const unsigned j  = (unsigned)perm[(b << 10) + pi];  // source patch

    // Source pixel coordinates: same (r, c-within-patch), different patch
    const unsigned iny = ((j >> 5) << 4) | (y & 15u);
    const unsigned inx = ((j & 31u) << 4) | (x & 15u);

    const unsigned base   = b * kC * kPlane;             // start of image b
    const unsigned inOff  = base + iny * kW + inx;       // channel 0 src
    const unsigned outOff = base + y   * kW + x;         // channel 0 dst

    // Issue all three channel loads (3x global_load_b128 in flight), then
    // the three coalesced stores (3x global_store_b128).
    const float4 v0 = *reinterpret_cast<const float4*>(img + inOff);
    const float4 v1 = *reinterpret_cast<const float4*>(img + inOff + kPlane);
    const float4 v2 = *reinterpret_cast<const float4*>(img + inOff + 2u * kPlane);

    *reinterpret_cast<float4*>(out + outOff)              = v0;
    *reinterpret_cast<float4*>(out + outOff + kPlane)     = v1;
    *reinterpret_cast<float4*>(out + outOff + 2u * kPlane) = v2;
}

extern "C" void kernel_launch(void* const* d_in, const int* in_sizes, int n_in,
                              void* d_out, int out_size, void* d_ws, size_t ws_size,
                              hipStream_t stream) {
    const float* img  = (const float*)d_in[0];   // [64, 3, 512, 512] fp32
    const int*   perm = (const int*)d_in[1];     // [64, 1024] int32
    float*       out  = (float*)d_out;           // [64, 3, 512, 512] fp32

    const unsigned totalThreads = kB * kThreadsPerImage;   // 4,194,304
    const unsigned blocks = totalThreads / 256u;           // 16,384

    patch_shuffle_kernel<<<blocks, 256, 0, stream>>>(img, perm, out);
}